// Loss4PixelReconstruction_85203561218674
// MI455X (gfx1250) — compile-verified
//
#include <hip/hip_runtime.h>
#include <math.h>

// Problem constants (fixed by the reference)
#define KF    11
#define PAD   5            // KF/2
#define Hd    256
#define Wd    256
#define Nn    4
#define Cc    3
#define HW    (Hd * Wd)            // 65536
#define TAPS  (KF * KF)            // 121
#define TCOLS (Wd + 2 * PAD)       // 266
#define EPS2  1e-6f                // (1e-3)^2
#define TOTAL_ELEMS (Nn * Cc * HW) // 786432

typedef float v2f __attribute__((ext_vector_type(2)));
typedef float v8f __attribute__((ext_vector_type(8)));

// Wave32 sum-reduction using the CDNA5 matrix pipe:
//   A (16x4 f32) carries the 32 lane partials (lane m -> A[m][0], lane m+16 -> A[m][2],
//   K=1/K=3 slots zeroed), B = ones(4x16).  D[m][n] = rowsum_m = s_m + s_{m+16},
//   replicated across columns.  Each lane then folds its 8 D registers
//   (rowsums m..m+7 for its half) and one shfl_xor(16) completes the wave sum.
// Requires EXEC == all ones: call with a full, convergent wave.
__device__ __forceinline__ float wave_sum_wmma(float s) {
    v2f a; a[0] = s;    a[1] = 0.0f;
    v2f b; b[0] = 1.0f; b[1] = 1.0f;
    v8f c = {};
    c = __builtin_amdgcn_wmma_f32_16x16x4_f32(
            /*neg_a=*/false, a, /*neg_b=*/false, b,
            /*c_mod=*/(short)0, c, /*reuse_a=*/false, /*reuse_b=*/false);
    float p = ((c[0] + c[1]) + (c[2] + c[3])) + ((c[4] + c[5]) + (c[6] + c[7]));
    p += __shfl_xor(p, 16, 32);   // combine the two half-wave rowsum groups
    return p;                     // full wave sum, uniform across lanes
}

// One block per (n, h) output row; thread t owns pixel w = t.
__global__ __launch_bounds__(256)
void fdn_charbonnier_kernel(const float* __restrict__ img1,
                            const float* __restrict__ img2,
                            const float* __restrict__ filt,
                            float* __restrict__ block_partial) {
    // Channel-interleaved padded tile: 11 rows x 266 cols x float4 (xyz = RGB).
    __shared__ float4 tile[KF * TCOLS];           // 46,816 bytes
    __shared__ float  wsum[8];

    const int n   = blockIdx.x >> 8;              // 0..3
    const int h   = blockIdx.x & 255;             // 0..255
    const int tid = threadIdx.x;

    // ---- Stage image1 rows [h-5, h+5] with zero padding (coalesced along col) ----
    const size_t i1base = (size_t)n * Cc * HW;
    for (int idx = tid; idx < KF * TCOLS; idx += 256) {
        const int r   = idx / TCOLS;
        const int col = idx - r * TCOLS;
        const int hh  = h - PAD + r;
        const int ww  = col - PAD;
        float4 v = make_float4(0.0f, 0.0f, 0.0f, 0.0f);
        if ((unsigned)hh < (unsigned)Hd && (unsigned)ww < (unsigned)Wd) {
            const size_t p = i1base + (size_t)hh * Wd + (size_t)ww;
            v.x = img1[p];
            v.y = img1[p + HW];
            v.z = img1[p + 2 * HW];
        }
        tile[idx] = v;
    }
    __syncthreads();

    // ---- Stream 121 filter taps (non-temporal: single-use, 121 MB total) ----
    const int w = tid;
    const float* fp = filt + (size_t)n * TAPS * HW + (size_t)h * Wd + (size_t)w;

    float a0 = 0.0f, a1 = 0.0f, a2 = 0.0f;
    for (int kh = 0; kh < KF; ++kh) {
        if (kh + 2 < KF)   // prefetch two tap-rows ahead of the NT stream
            __builtin_prefetch(fp + (size_t)((kh + 2) * KF) * HW, 0, 0);
        const float4* row = &tile[kh * TCOLS + w];
#pragma unroll
        for (int kw = 0; kw < KF; ++kw) {
            const float  f  = __builtin_nontemporal_load(fp + (size_t)(kh * KF + kw) * HW);
            const float4 px = row[kw];            // ds_load_b128, all 3 channels
            a0 = fmaf(px.x, f, a0);
            a1 = fmaf(px.y, f, a1);
            a2 = fmaf(px.z, f, a2);
        }
    }

    // ---- Charbonnier over the 3 channels ----
    const size_t ob = i1base + (size_t)h * Wd + (size_t)w;
    const float d0 = a0 - img2[ob];
    const float d1 = a1 - img2[ob + HW];
    const float d2 = a2 - img2[ob + 2 * HW];
    float s = sqrtf(fmaf(d0, d0, EPS2))
            + sqrtf(fmaf(d1, d1, EPS2))
            + sqrtf(fmaf(d2, d2, EPS2));

    // ---- Deterministic block reduction: WMMA wave sum, then LDS fold ----
    const int lane = tid & 31;
    const int wid  = tid >> 5;
    const float wv = wave_sum_wmma(s);
    if (lane == 0) wsum[wid] = wv;
    __syncthreads();
    if (tid == 0) {
        float t = 0.0f;
#pragma unroll
        for (int i = 0; i < 8; ++i) t += wsum[i];
        block_partial[blockIdx.x] = t;
    }
}

// Single-block, fixed-order final reduction of the 1024 block partials.
__global__ __launch_bounds__(256)
void final_reduce_kernel(const float* __restrict__ block_partial,
                         float* __restrict__ out) {
    __shared__ float wsum[8];
    const int tid = threadIdx.x;
    float s = block_partial[tid]
            + block_partial[tid + 256]
            + block_partial[tid + 512]
            + block_partial[tid + 768];
    const float wv = wave_sum_wmma(s);
    const int lane = tid & 31;
    const int wid  = tid >> 5;
    if (lane == 0) wsum[wid] = wv;
    __syncthreads();
    if (tid == 0) {
        float t = 0.0f;
#pragma unroll
        for (int i = 0; i < 8; ++i) t += wsum[i];
        out[0] = t * (1.0f / (float)TOTAL_ELEMS);
    }
}

extern "C" void kernel_launch(void* const* d_in, const int* in_sizes, int n_in,
                              void* d_out, int out_size, void* d_ws, size_t ws_size,
                              hipStream_t stream) {
    (void)in_sizes; (void)n_in; (void)out_size; (void)ws_size;
    const float* img1 = (const float*)d_in[0];   // (N,C,H,W) fp32
    const float* img2 = (const float*)d_in[1];   // (N,C,H,W) fp32
    const float* filt = (const float*)d_in[2];   // (N,K*K,H,W) fp32
    float* partial = (float*)d_ws;               // 1024 floats, fully overwritten
    float* out     = (float*)d_out;              // 1 float

    fdn_charbonnier_kernel<<<Nn * Hd, 256, 0, stream>>>(img1, img2, filt, partial);
    final_reduce_kernel<<<1, 256, 0, stream>>>(partial, out);
}